// DSA_47287589929772
// MI455X (gfx1250) — compile-verified
//
#include <hip/hip_runtime.h>

typedef float v2f __attribute__((ext_vector_type(2)));
typedef float v4f __attribute__((ext_vector_type(4)));
typedef float v8f __attribute__((ext_vector_type(8)));

#define C_DIM 128
#define LDS_PAD 4
#define GEMM_GRID 1024
#define POOL_GRID 2048
#define MAXG 2048
#define BN_EPS 1e-5f

__device__ __forceinline__ v4f vmax4(v4f a, v4f b) {
    v4f r;
    r.x = fmaxf(a.x, b.x); r.y = fmaxf(a.y, b.y);
    r.z = fmaxf(a.z, b.z); r.w = fmaxf(a.w, b.w);
    return r;
}

// ---------------------------------------------------------------------------
// GEMM: Out[n][j] = sum_k T(A[n][k]) * W[k][j] + bias[j]
// T: mode 0 = identity, mode 1 = affine (scale/shift), mode 2 = relu(affine)
// Also emits per-block column sum / sumsq partials for the following BatchNorm.
// Each block: 16-row tiles (grid-stride); wave w owns columns [16w, 16w+16).
// Uses v_wmma_f32_16x16x4_f32 (fp32 WMMA), K swept 128/4 = 32 steps.
// ---------------------------------------------------------------------------
__global__ __launch_bounds__(256) void gemm_bn_kernel(
    const float* __restrict__ A, const float* __restrict__ W,
    const float* __restrict__ bias,
    const float* __restrict__ scale, const float* __restrict__ shift,
    float* __restrict__ Out,
    float* __restrict__ psum, float* __restrict__ psq,
    int N, int mode)
{
    __shared__ float ldsA[16][C_DIM + LDS_PAD];

    const int tid  = threadIdx.x;
    const int wave = tid >> 5;          // 0..7
    const int lane = tid & 31;
    const int half = lane >> 4;         // 0 or 1 (half-wave, selects K pair)
    const int l16  = lane & 15;         // row (A) / column-in-tile (B,D)
    const int n0   = wave << 4;         // column tile base
    const int col  = n0 + l16;
    const float bcol = bias[col];

    float sumAcc = 0.f, sqAcc = 0.f;
    const int numTiles = N >> 4;

    for (int tile = blockIdx.x; tile < numTiles; tile += gridDim.x) {
        const int rowBase = tile << 4;
        __syncthreads();
        // Cooperative coalesced load of the 16x128 A tile (+ fused BN/ReLU)
        for (int i = tid; i < 16 * 32; i += 256) {
            const int r  = i >> 5;
            const int c4 = (i & 31) << 2;
            v4f v = *(const v4f*)(A + (size_t)(rowBase + r) * C_DIM + c4);
            if (mode != 0) {
                v.x = v.x * scale[c4 + 0] + shift[c4 + 0];
                v.y = v.y * scale[c4 + 1] + shift[c4 + 1];
                v.z = v.z * scale[c4 + 2] + shift[c4 + 2];
                v.w = v.w * scale[c4 + 3] + shift[c4 + 3];
                if (mode == 2) {
                    v.x = fmaxf(v.x, 0.f); v.y = fmaxf(v.y, 0.f);
                    v.z = fmaxf(v.z, 0.f); v.w = fmaxf(v.w, 0.f);
                }
            }
            *(v4f*)&ldsA[r][c4] = v;
        }
        __syncthreads();

        v8f acc = {};
        #pragma unroll
        for (int k0 = 0; k0 < C_DIM; k0 += 4) {
            const int kk = k0 + (half << 1);
            // A frag 16x4: lanes 0-15 K={k0,k0+1}, lanes 16-31 K={k0+2,k0+3}
            v2f a, b;
            a.x = ldsA[l16][kk];
            a.y = ldsA[l16][kk + 1];
            // B frag 4x16: same K striping, N across lanes
            b.x = W[kk * C_DIM + col];
            b.y = W[(kk + 1) * C_DIM + col];
            acc = __builtin_amdgcn_wmma_f32_16x16x4_f32(
                false, a, false, b, (short)0, acc, false, false);
        }

        // D layout: VGPR r = row (half?8+r:r), col = n0 + l16
        #pragma unroll
        for (int r = 0; r < 8; ++r) {
            const int row = (half ? 8 : 0) + r;
            const float v = acc[r] + bcol;
            Out[(size_t)(rowBase + row) * C_DIM + col] = v;
            sumAcc += v;
            sqAcc  += v * v;
        }
    }

    // combine the two half-waves (same column) and write per-block partials
    sumAcc += __shfl_xor(sumAcc, 16, 32);
    sqAcc  += __shfl_xor(sqAcc, 16, 32);
    if (half == 0) {
        psum[blockIdx.x * C_DIM + col] = sumAcc;
        psq [blockIdx.x * C_DIM + col] = sqAcc;
    }
}

// ---------------------------------------------------------------------------
// Fold partial stats into per-channel scale/shift:
//   scale = gamma * rsqrt(var+eps), shift = beta - mean*scale
// ---------------------------------------------------------------------------
__global__ __launch_bounds__(128) void bn_finalize_kernel(
    const float* __restrict__ psum, const float* __restrict__ psq, int G,
    const float* __restrict__ gamma, const float* __restrict__ beta,
    float* __restrict__ scale, float* __restrict__ shift, float invN)
{
    const int c = threadIdx.x;
    float s = 0.f, q = 0.f;
    for (int g = 0; g < G; ++g) {
        s += psum[g * C_DIM + c];
        q += psq [g * C_DIM + c];
    }
    const float m  = s * invN;
    const float v  = fmaxf(q * invN - m * m, 0.f);
    const float sc = gamma[c] * rsqrtf(v + BN_EPS);
    scale[c] = sc;
    shift[c] = beta[c] - m * sc;
}

// ---------------------------------------------------------------------------
// h1 = relu(t1*scale + shift), elementwise (in-place OK)
// ---------------------------------------------------------------------------
__global__ __launch_bounds__(256) void affine_relu_kernel(
    const float* __restrict__ in,
    const float* __restrict__ scale, const float* __restrict__ shift,
    float* __restrict__ out, int total4)
{
    const int i = blockIdx.x * 256 + threadIdx.x;
    if (i >= total4) return;
    const int c4 = (i & 31) << 2;
    v4f x = *(const v4f*)(in + (size_t)i * 4);
    x.x = fmaxf(x.x * scale[c4 + 0] + shift[c4 + 0], 0.f);
    x.y = fmaxf(x.y * scale[c4 + 1] + shift[c4 + 1], 0.f);
    x.z = fmaxf(x.z * scale[c4 + 2] + shift[c4 + 2], 0.f);
    x.w = fmaxf(x.w * scale[c4 + 3] + shift[c4 + 3], 0.f);
    *(v4f*)(out + (size_t)i * 4) = x;
}

// ---------------------------------------------------------------------------
// pooled[n][c] = max_k ( pe[n][k][c] + h1[knn[n][k]][c] ), plus BN2 partials.
// 32 threads per point (float4 over C=128). pe is streamed non-temporally
// (one-shot 819MB) so the L2 stays warm for the h1 gather (51MB, L2-resident).
// ---------------------------------------------------------------------------
__global__ __launch_bounds__(256) void gather_pool_kernel(
    const float* __restrict__ pe, const float* __restrict__ h1,
    const int* __restrict__ knn, float* __restrict__ pooled,
    float* __restrict__ psum, float* __restrict__ psq, int N)
{
    __shared__ v4f s1[256];
    __shared__ v4f s2[256];
    const int t  = threadIdx.x;
    const int g  = t >> 5;           // point group 0..7
    const int c4 = (t & 31) << 2;    // channel base

    v4f sum = {0.f, 0.f, 0.f, 0.f};
    v4f sq  = {0.f, 0.f, 0.f, 0.f};

    for (int n = blockIdx.x * 8 + g; n < N; n += gridDim.x * 8) {
        v4f m = {-3.402823466e38f, -3.402823466e38f,
                 -3.402823466e38f, -3.402823466e38f};
        const int* idx = knn + (size_t)n * 16;
        const float* peRow = pe + (size_t)n * (16 * C_DIM) + c4;
        #pragma unroll
        for (int k = 0; k < 16; ++k) {
            const int j = idx[k];
            v4f pv = __builtin_nontemporal_load((const v4f*)(peRow + (size_t)k * C_DIM));
            v4f hv = *(const v4f*)(h1 + (size_t)j * C_DIM + c4);
            m = vmax4(m, pv + hv);
        }
        *(v4f*)(pooled + (size_t)n * C_DIM + c4) = m;
        sum += m;
        sq  += m * m;
    }

    s1[t] = sum; s2[t] = sq;
    __syncthreads();
    for (int s = 128; s >= 32; s >>= 1) {
        if (t < s) { s1[t] += s1[t + s]; s2[t] += s2[t + s]; }
        __syncthreads();
    }
    if (t < 32) {
        *(v4f*)(psum + blockIdx.x * C_DIM + t * 4) = s1[t];
        *(v4f*)(psq  + blockIdx.x * C_DIM + t * 4) = s2[t];
    }
}

// ---------------------------------------------------------------------------
// out = relu(f + t3*scale + shift)   (t3 lives in `out` already; in-place)
// ---------------------------------------------------------------------------
__global__ __launch_bounds__(256) void residual_kernel(
    const float* __restrict__ f, const float* __restrict__ t3,
    const float* __restrict__ scale, const float* __restrict__ shift,
    float* __restrict__ out, int total4)
{
    const int i = blockIdx.x * 256 + threadIdx.x;
    if (i >= total4) return;
    const int c4 = (i & 31) << 2;
    v4f x = *(const v4f*)(t3 + (size_t)i * 4);
    v4f idv = *(const v4f*)(f + (size_t)i * 4);
    x.x = fmaxf(idv.x + x.x * scale[c4 + 0] + shift[c4 + 0], 0.f);
    x.y = fmaxf(idv.y + x.y * scale[c4 + 1] + shift[c4 + 1], 0.f);
    x.z = fmaxf(idv.z + x.z * scale[c4 + 2] + shift[c4 + 2], 0.f);
    x.w = fmaxf(idv.w + x.w * scale[c4 + 3] + shift[c4 + 3], 0.f);
    *(v4f*)(out + (size_t)i * 4) = x;
}

extern "C" void kernel_launch(void* const* d_in, const int* in_sizes, int n_in,
                              void* d_out, int out_size, void* d_ws, size_t ws_size,
                              hipStream_t stream)
{
    // setup_inputs order: p, f, pe, knn_index, W_pre, b_pre, g1, be1, g2, be2,
    //                     W_f1, b_f1, g3, be3, W_f2, b_f2, g4, be4
    const float* f    = (const float*)d_in[1];
    const float* pe   = (const float*)d_in[2];
    const int*   knn  = (const int*)  d_in[3];
    const float* Wpre = (const float*)d_in[4];
    const float* bpre = (const float*)d_in[5];
    const float* g1   = (const float*)d_in[6];  const float* be1 = (const float*)d_in[7];
    const float* g2   = (const float*)d_in[8];  const float* be2 = (const float*)d_in[9];
    const float* Wf1  = (const float*)d_in[10]; const float* bf1 = (const float*)d_in[11];
    const float* g3   = (const float*)d_in[12]; const float* be3 = (const float*)d_in[13];
    const float* Wf2  = (const float*)d_in[14]; const float* bf2 = (const float*)d_in[15];
    const float* g4   = (const float*)d_in[16]; const float* be4 = (const float*)d_in[17];

    const int N = in_sizes[1] / C_DIM;
    const float invN = 1.0f / (float)N;

    char* ws = (char*)d_ws;
    const size_t NC  = (size_t)N * C_DIM * sizeof(float);
    const size_t PCH = (size_t)MAXG * C_DIM * sizeof(float);
    float* bufA = (float*)(ws);                  // t1 -> h1 -> t2
    float* bufB = (float*)(ws + NC);             // pooled
    float* psum = (float*)(ws + 2 * NC);
    float* psq  = (float*)(ws + 2 * NC + PCH);
    float* prm  = (float*)(ws + 2 * NC + 2 * PCH);
    float* sc1 = prm + 0 * C_DIM; float* sh1 = prm + 1 * C_DIM;
    float* sc2 = prm + 2 * C_DIM; float* sh2 = prm + 3 * C_DIM;
    float* sc3 = prm + 4 * C_DIM; float* sh3 = prm + 5 * C_DIM;
    float* sc4 = prm + 6 * C_DIM; float* sh4 = prm + 7 * C_DIM;
    float* out = (float*)d_out;

    const int total4   = N * (C_DIM / 4);
    const int ewBlocks = (total4 + 255) / 256;

    // 1) t1 = f @ W_pre + b_pre ; BN1 partials
    gemm_bn_kernel<<<GEMM_GRID, 256, 0, stream>>>(f, Wpre, bpre, nullptr, nullptr,
                                                  bufA, psum, psq, N, 0);
    bn_finalize_kernel<<<1, 128, 0, stream>>>(psum, psq, GEMM_GRID, g1, be1, sc1, sh1, invN);
    // 2) h1 = relu(BN1(t1))  (in place)
    affine_relu_kernel<<<ewBlocks, 256, 0, stream>>>(bufA, sc1, sh1, bufA, total4);
    // 3) pooled = max_k(pe + h1[knn]) ; BN2 partials
    gather_pool_kernel<<<POOL_GRID, 256, 0, stream>>>(pe, bufA, knn, bufB, psum, psq, N);
    bn_finalize_kernel<<<1, 128, 0, stream>>>(psum, psq, POOL_GRID, g2, be2, sc2, sh2, invN);
    // 4) t2 = BN2(pooled) @ W_f1 + b_f1 ; BN3 partials
    gemm_bn_kernel<<<GEMM_GRID, 256, 0, stream>>>(bufB, Wf1, bf1, sc2, sh2,
                                                  bufA, psum, psq, N, 1);
    bn_finalize_kernel<<<1, 128, 0, stream>>>(psum, psq, GEMM_GRID, g3, be3, sc3, sh3, invN);
    // 5) t3 = relu(BN3(t2)) @ W_f2 + b_f2 ; BN4 partials (t3 -> d_out)
    gemm_bn_kernel<<<GEMM_GRID, 256, 0, stream>>>(bufA, Wf2, bf2, sc3, sh3,
                                                  out, psum, psq, N, 2);
    bn_finalize_kernel<<<1, 128, 0, stream>>>(psum, psq, GEMM_GRID, g4, be4, sc4, sh4, invN);
    // 6) out = relu(f + BN4(t3))
    residual_kernel<<<ewBlocks, 256, 0, stream>>>(f, out, sc4, sh4, out, total4);
}